// PointNetFeaturePropagation_46119358824788
// MI455X (gfx1250) — compile-verified
//
#include <hip/hip_runtime.h>
#include <math.h>

typedef __attribute__((ext_vector_type(2))) float v2f;
typedef __attribute__((ext_vector_type(8))) float v8f;

#define B_    8
#define N_    4096
#define S_    1024
#define D1_   128
#define D2_   256
#define INCH  384
#define OUTCH 256
#define M_    (B_ * N_)   // 32768

__device__ __forceinline__ float gelu_exact(float x) {
    return 0.5f * x * (1.0f + erff(x * 0.70710678118654752440f));
}

__device__ __forceinline__ void top3_insert(float dd, int ii,
                                            float& d0, float& d1, float& d2,
                                            int& i0, int& i1, int& i2) {
    if (dd < d0)      { d2 = d1; i2 = i1; d1 = d0; i1 = i0; d0 = dd; i0 = ii; }
    else if (dd < d1) { d2 = d1; i2 = i1; d1 = dd; i1 = ii; }
    else if (dd < d2) { d2 = dd; i2 = ii; }
}

// ---------------------------------------------------------------------------
// K1: 3-NN inverse-distance interpolation + concat  -> X0 (M x 384)
// block = 128 threads (4 waves, wave32); wave handles one query point.
// ---------------------------------------------------------------------------
__global__ __launch_bounds__(128) void knn_interp_kernel(
    const float* __restrict__ xyz1,    // (B,N,3)
    const float* __restrict__ xyz2,    // (B,S,3)
    const float* __restrict__ points1, // (B,N,128)
    const float* __restrict__ points2, // (B,S,256)
    float* __restrict__ X0)            // (M,384)
{
    __shared__ float sx[S_ * 3];
    const int b   = blockIdx.y;
    const int tid = threadIdx.x;

    const float* p2 = xyz2 + (size_t)b * S_ * 3;
    for (int i = tid; i < S_ * 3; i += 128) sx[i] = p2[i];
    __syncthreads();

    const int wave = tid >> 5;
    const int lane = tid & 31;
    const int n    = blockIdx.x * 4 + wave;
    const int m    = b * N_ + n;

    const float* q = xyz1 + (size_t)m * 3;
    const float qx = q[0], qy = q[1], qz = q[2];

    float d0 = 1e30f, d1 = 1e30f, d2 = 1e30f;
    int   i0 = 0,     i1 = 0,     i2 = 0;

    for (int s = lane; s < S_; s += 32) {
        float dx = qx - sx[s * 3 + 0];
        float dy = qy - sx[s * 3 + 1];
        float dz = qz - sx[s * 3 + 2];
        float dd = dx * dx + dy * dy + dz * dz;
        top3_insert(dd, s, d0, d1, d2, i0, i1, i2);
    }
    // wave32 shuffle-tree merge of per-lane top-3 lists
    for (int off = 16; off > 0; off >>= 1) {
        float e0 = __shfl_down(d0, off, 32); int j0 = __shfl_down(i0, off, 32);
        float e1 = __shfl_down(d1, off, 32); int j1 = __shfl_down(i1, off, 32);
        float e2 = __shfl_down(d2, off, 32); int j2 = __shfl_down(i2, off, 32);
        top3_insert(e0, j0, d0, d1, d2, i0, i1, i2);
        top3_insert(e1, j1, d0, d1, d2, i0, i1, i2);
        top3_insert(e2, j2, d0, d1, d2, i0, i1, i2);
    }
    // broadcast winners from lane 0
    d0 = __shfl(d0, 0, 32); d1 = __shfl(d1, 0, 32); d2 = __shfl(d2, 0, 32);
    i0 = __shfl(i0, 0, 32); i1 = __shfl(i1, 0, 32); i2 = __shfl(i2, 0, 32);

    const float r0 = 1.0f / (d0 + 1e-8f);
    const float r1 = 1.0f / (d1 + 1e-8f);
    const float r2 = 1.0f / (d2 + 1e-8f);
    const float rs = 1.0f / (r0 + r1 + r2);
    const float w0 = r0 * rs, w1 = r1 * rs, w2 = r2 * rs;

    float* xrow = X0 + (size_t)m * INCH;

    // copy points1 (128 ch -> 4 per lane, vectorized)
    const float4 p1v = *(const float4*)(points1 + (size_t)m * D1_ + lane * 4);
    *(float4*)(xrow + lane * 4) = p1v;

    // interpolate points2 (256 ch -> 8 per lane)
    const float* g0 = points2 + ((size_t)b * S_ + i0) * D2_;
    const float* g1 = points2 + ((size_t)b * S_ + i1) * D2_;
    const float* g2 = points2 + ((size_t)b * S_ + i2) * D2_;
    const int cb = lane * 8;
#pragma unroll
    for (int c = 0; c < 8; c += 4) {
        float4 a0 = *(const float4*)(g0 + cb + c);
        float4 a1 = *(const float4*)(g1 + cb + c);
        float4 a2 = *(const float4*)(g2 + cb + c);
        float4 o;
        o.x = w0 * a0.x + w1 * a1.x + w2 * a2.x;
        o.y = w0 * a0.y + w1 * a1.y + w2 * a2.y;
        o.z = w0 * a0.z + w1 * a1.z + w2 * a2.z;
        o.w = w0 * a0.w + w1 * a1.w + w2 * a2.w;
        *(float4*)(xrow + D1_ + cb + c) = o;
    }
}

// ---------------------------------------------------------------------------
// GEMM:  Z(M x 256) = X(M x K) * W(256 x K)^T + bias    (fp32 WMMA 16x16x4)
// block = 256 threads = 8 waves (4 M x 2 N); block tile 64x64; K-step 32.
// LDS double-buffered; A stored with even stride and B stored K-paired so
// every wmma fragment is a single aligned ds_load_b64 (no repacking movs).
// ---------------------------------------------------------------------------
#define MBLK   64
#define NBLK   64
#define KSTEP  32
#define AS_LD  36   // floats; 36*m mod 64 distinct per half-wave -> no bank conflicts
#define BS2_LD 80   // float2; 2*80 mod 64 == 32 -> halves of wave hit disjoint banks

__global__ __launch_bounds__(256) void gemm_wmma_kernel(
    const float* __restrict__ X,    // M x K
    const float* __restrict__ W,    // OUTCH x K
    const float* __restrict__ bias, // OUTCH
    float* __restrict__ Z,          // M x OUTCH
    int M, int K)
{
    __shared__ float As[2][MBLK * AS_LD];            // 2 x 9216 B
    __shared__ v2f   Bs[2][(KSTEP / 2) * BS2_LD];    // 2 x 10240 B

    const int tid  = threadIdx.x;
    const int lane = tid & 31;
    const int wave = tid >> 5;
    const int wm   = wave & 3;   // 4 waves along M
    const int wn   = wave >> 2;  // 2 waves along N
    const int row0 = blockIdx.x * MBLK;
    const int col0 = blockIdx.y * NBLK;

    const int ml = lane & 15;          // fragment row/col within 16
    const int kp = (lane >> 4) << 1;   // K-pair select (ISA 16x4 fp32 A layout)
    const int kh = lane >> 4;          // paired-K row select for B

    v8f acc0 = {};
    v8f acc1 = {};

    const int sm = tid >> 2;           // staging: row / out-channel index (0..63)
    const int sk = (tid & 3) * 8;      // staging: 8 consecutive K elements

    auto stage = [&](int k0, int buf) {
        // A tile: As[m][k] = X[row0+m][k0+k]
        const float* srcA = X + (size_t)(row0 + sm) * K + k0 + sk;
        float4 a0 = *(const float4*)(srcA);
        float4 a1 = *(const float4*)(srcA + 4);
        float* adst = &As[buf][sm * AS_LD + sk];
        *(float4*)(adst)     = a0;
        *(float4*)(adst + 4) = a1;
        // B tile, K-paired + transposed: Bs[k/2][n] = {W[n][k0+k], W[n][k0+k+1]}
        const float* srcB = W + (size_t)(col0 + sm) * K + k0 + sk;
        float4 w0 = *(const float4*)(srcB);
        float4 w1 = *(const float4*)(srcB + 4);
        const int khb = sk >> 1;
        v2f p;
        p.x = w0.x; p.y = w0.y; Bs[buf][(khb + 0) * BS2_LD + sm] = p;
        p.x = w0.z; p.y = w0.w; Bs[buf][(khb + 1) * BS2_LD + sm] = p;
        p.x = w1.x; p.y = w1.y; Bs[buf][(khb + 2) * BS2_LD + sm] = p;
        p.x = w1.z; p.y = w1.w; Bs[buf][(khb + 3) * BS2_LD + sm] = p;
    };

    stage(0, 0);
    int buf = 0;

    for (int k0 = 0; k0 < K; k0 += KSTEP) {
        __syncthreads();   // staging of `buf` complete; previous compute done

        if (k0 + KSTEP < K) {
            stage(k0 + KSTEP, buf ^ 1);          // overlap with compute below
            if (k0 + 2 * KSTEP < K) {            // keep L2 ahead (global_prefetch_b8)
                __builtin_prefetch(X + (size_t)(row0 + sm) * K + k0 + 2 * KSTEP + sk, 0, 1);
                __builtin_prefetch(W + (size_t)(col0 + sm) * K + k0 + 2 * KSTEP + sk, 0, 1);
            }
        }

        const float* Ab = &As[buf][0];
        const v2f*   Bb = &Bs[buf][0];
#pragma unroll
        for (int kk = 0; kk < KSTEP; kk += 4) {
            // A fragment (16x4 fp32): one aligned ds_load_b64 per lane
            v2f a = *(const v2f*)(Ab + (wm * 16 + ml) * AS_LD + kk + kp);
            // B fragments (4x16 fp32): paired layout -> one ds_load_b64 each
            const int krow = ((kk >> 1) + kh) * BS2_LD + wn * 32 + ml;
            v2f b0 = Bb[krow];
            v2f b1 = Bb[krow + 16];
            acc0 = __builtin_amdgcn_wmma_f32_16x16x4_f32(false, a, false, b0,
                                                         (short)0, acc0, false, false);
            acc1 = __builtin_amdgcn_wmma_f32_16x16x4_f32(false, a, false, b1,
                                                         (short)0, acc1, false, false);
        }
        buf ^= 1;
    }

    // epilogue: C/D layout -> lane l holds rows {r + 8*(l>>4)}, col = l&15
    const int rbase = row0 + wm * 16 + (lane >> 4) * 8;
    const int cbase = col0 + wn * 32 + ml;
    const float bv0 = bias[cbase];
    const float bv1 = bias[cbase + 16];
#pragma unroll
    for (int r = 0; r < 8; ++r) {
        Z[(size_t)(rbase + r) * OUTCH + cbase]      = acc0[r] + bv0;
        Z[(size_t)(rbase + r) * OUTCH + cbase + 16] = acc1[r] + bv1;
    }
}

// ---------------------------------------------------------------------------
// BN statistics: deterministic two-phase reduction (no atomics)
// ---------------------------------------------------------------------------
#define STAT_BLOCKS 128

__global__ __launch_bounds__(256) void stats_partial_kernel(
    const float* __restrict__ Z, float* __restrict__ partial, int M)
{
    const int c    = threadIdx.x;           // channel
    const int rows = M / STAT_BLOCKS;
    const int r0   = blockIdx.x * rows;
    float s = 0.0f, q = 0.0f;
    for (int r = 0; r < rows; ++r) {        // coalesced: 256 lanes read one row
        float v = Z[(size_t)(r0 + r) * OUTCH + c];
        s += v;
        q += v * v;
    }
    partial[blockIdx.x * (2 * OUTCH) + c]         = s;
    partial[blockIdx.x * (2 * OUTCH) + OUTCH + c] = q;
}

__global__ __launch_bounds__(256) void stats_finalize_kernel(
    const float* __restrict__ partial,
    const float* __restrict__ g, const float* __restrict__ bt,
    float* __restrict__ ss, int M)
{
    const int c = threadIdx.x;
    float s = 0.0f, q = 0.0f;
    for (int i = 0; i < STAT_BLOCKS; ++i) {
        s += partial[i * (2 * OUTCH) + c];
        q += partial[i * (2 * OUTCH) + OUTCH + c];
    }
    const float mean  = s / (float)M;
    const float var   = q / (float)M - mean * mean;
    const float scale = g[c] * rsqrtf(var + 1e-5f);
    ss[c]         = scale;
    ss[OUTCH + c] = bt[c] - mean * scale;
}

__global__ __launch_bounds__(256) void bn_gelu_kernel(
    const float* __restrict__ Z, const float* __restrict__ ss,
    float* __restrict__ Xo, int total)
{
    int i = blockIdx.x * 256 + threadIdx.x;
    if (i < total) {
        int c = i & (OUTCH - 1);
        Xo[i] = gelu_exact(Z[i] * ss[c] + ss[OUTCH + c]);
    }
}

__global__ __launch_bounds__(256) void bn_residual_gelu_kernel(
    const float* __restrict__ Z, const float* __restrict__ ss,
    const float* __restrict__ X1, float* __restrict__ out, int total)
{
    int i = blockIdx.x * 256 + threadIdx.x;
    if (i < total) {
        int c = i & (OUTCH - 1);
        out[i] = gelu_exact(Z[i] * ss[c] + ss[OUTCH + c] + X1[i]);
    }
}

// ---------------------------------------------------------------------------
extern "C" void kernel_launch(void* const* d_in, const int* in_sizes, int n_in,
                              void* d_out, int out_size, void* d_ws, size_t ws_size,
                              hipStream_t stream)
{
    const float* xyz1    = (const float*)d_in[0];
    const float* xyz2    = (const float*)d_in[1];
    const float* points1 = (const float*)d_in[2];
    const float* points2 = (const float*)d_in[3];
    const float* W_fuse  = (const float*)d_in[4];
    const float* b_fuse  = (const float*)d_in[5];
    const float* g_fuse  = (const float*)d_in[6];
    const float* bt_fuse = (const float*)d_in[7];
    const float* W1      = (const float*)d_in[8];
    const float* b1      = (const float*)d_in[9];
    const float* g1      = (const float*)d_in[10];
    const float* bt1     = (const float*)d_in[11];
    const float* W2      = (const float*)d_in[12];
    const float* b2      = (const float*)d_in[13];
    const float* g2      = (const float*)d_in[14];
    const float* bt2     = (const float*)d_in[15];
    float*       out     = (float*)d_out;

    // workspace layout (bytes)
    char* ws = (char*)d_ws;
    const size_t szA = (size_t)M_ * INCH  * 4;  // 50.3 MB : X0, later X1
    const size_t szC = (size_t)M_ * OUTCH * 4;  // 33.5 MB
    float* regA    = (float*)(ws);                         // X0 -> X1
    float* regB    = (float*)(ws + szA);                   // Z1 -> X2
    float* regC    = (float*)(ws + szA + szC);             // Z2 -> Z3
    float* partial = (float*)(ws + szA + 2 * szC);         // 128*512 floats
    float* ssbuf   = (float*)(ws + szA + 2 * szC + (size_t)STAT_BLOCKS * 2 * OUTCH * 4);

    const int total = M_ * OUTCH;
    const dim3 gemm_grid(M_ / MBLK, OUTCH / NBLK);

    // Stage 0: KNN interpolation + feature concat -> X0
    knn_interp_kernel<<<dim3(N_ / 4, B_), 128, 0, stream>>>(
        xyz1, xyz2, points1, points2, regA);

    // Stage 1: fuse conv + BN + GELU
    gemm_wmma_kernel<<<gemm_grid, 256, 0, stream>>>(regA, W_fuse, b_fuse, regB, M_, INCH);
    stats_partial_kernel<<<STAT_BLOCKS, 256, 0, stream>>>(regB, partial, M_);
    stats_finalize_kernel<<<1, 256, 0, stream>>>(partial, g_fuse, bt_fuse, ssbuf, M_);
    bn_gelu_kernel<<<total / 256, 256, 0, stream>>>(regB, ssbuf, regA, total); // X1 -> regA

    // Stage 2: conv1 + BN + GELU
    gemm_wmma_kernel<<<gemm_grid, 256, 0, stream>>>(regA, W1, b1, regC, M_, OUTCH);
    stats_partial_kernel<<<STAT_BLOCKS, 256, 0, stream>>>(regC, partial, M_);
    stats_finalize_kernel<<<1, 256, 0, stream>>>(partial, g1, bt1, ssbuf, M_);
    bn_gelu_kernel<<<total / 256, 256, 0, stream>>>(regC, ssbuf, regB, total); // X2 -> regB

    // Stage 3: conv2 + BN + residual + GELU -> out
    gemm_wmma_kernel<<<gemm_grid, 256, 0, stream>>>(regB, W2, b2, regC, M_, OUTCH);
    stats_partial_kernel<<<STAT_BLOCKS, 256, 0, stream>>>(regC, partial, M_);
    stats_finalize_kernel<<<1, 256, 0, stream>>>(partial, g2, bt2, ssbuf, M_);
    bn_residual_gelu_kernel<<<total / 256, 256, 0, stream>>>(regC, ssbuf, regA, out, total);
}